// CRF_69750268887146
// MI455X (gfx1250) — compile-verified
//
#include <hip/hip_runtime.h>
#include <hip/hip_bf16.h>

#define LBL   128
#define TLEN  4096
#define STOPT 127
#define NEGV  (-10000.0f)

typedef __attribute__((ext_vector_type(2))) float v2f;
typedef __attribute__((ext_vector_type(8))) float v8f;

#if defined(__has_builtin)
#  if __has_builtin(__builtin_amdgcn_wmma_f32_16x16x4_f32)
#    define HAVE_WMMA_F32X4 1
#  endif
#endif

// ---------------------------------------------------------------------------
// Forward pass: sequential scan over T, one persistent workgroup (4 wave32).
//   thread j (0..127) owns label-column j.
//   Per step:
//     pass1 (VALU, exact f32): backpointer argmax_i(alphas[i]+trans[i][j]),
//                              plus maxA = max_i alphas[i] for free.
//     pass2 (WMMA f32 16x16x4): s[j] = sum_i exp(alphas[i]-maxA)*exp(trans[i][j])
//                              with E=exp(trans) weight-stationary in VGPRs.
//     alphas'[j] = maxA + log(s[j]) + emit[t][j]   (== reference logsumexp)
// ---------------------------------------------------------------------------
__global__ __launch_bounds__(128)
void crf_forward(const float* __restrict__ emit,
                 const float* __restrict__ trans,
                 float* __restrict__ out,          // out[0] = score
                 unsigned char* __restrict__ bt,   // T*L backpointers
                 int* __restrict__ bestLast)
{
    __shared__ float transLDS[LBL * LBL];   // 64 KB, row-major (i*L + j)
    __shared__ float alphas[LBL];
    __shared__ float eaLDS[LBL];            // exp(alphas - maxA)
    __shared__ float sBuf[LBL];

    const int tid  = threadIdx.x;           // 0..127 = column j
    const int lane = tid & 31;
    const int wave = tid >> 5;              // 0..3
    const int half = lane >> 4;             // selects K pair within a tile
    const int n    = lane & 15;             // N index within a 16-wide tile

    // Cooperative load of trans into LDS (conflict-free).
    for (int k = tid; k < LBL * LBL; k += 128) transLDS[k] = trans[k];
    alphas[tid] = (tid == STOPT) ? 0.0f : NEGV;
    __syncthreads();

#ifdef HAVE_WMMA_F32X4
    // Weight-stationary B tiles: E = exp(trans), 2 N-tiles x 32 K-tiles per wave.
    // B (4x16 f32) layout: VGPR0 = K{0|2} (lanes 0-15 | 16-31), VGPR1 = K{1|3}.
    v2f Breg[2][32];
    #pragma unroll
    for (int nt = 0; nt < 2; ++nt) {
        const int jb = (wave * 2 + nt) * 16 + n;
        #pragma unroll
        for (int kt = 0; kt < 32; ++kt) {
            const int r = kt * 4 + half * 2;
            v2f b;
            b.x = __expf(transLDS[(r + 0) * LBL + jb]);
            b.y = __expf(transLDS[(r + 1) * LBL + jb]);
            // refine with full-precision exp (cheap, init-only)
            b.x = expf(transLDS[(r + 0) * LBL + jb]);
            b.y = expf(transLDS[(r + 1) * LBL + jb]);
            Breg[nt][kt] = b;
        }
    }
#endif

    for (int t = 0; t < TLEN; ++t) {
        // ---- pass 1: argmax backpointers + global alpha max (exact f32) ----
        float mx   = -__builtin_inff();
        float maxA = -__builtin_inff();
        int   arg  = 0;
        #pragma unroll 8
        for (int i = 0; i < LBL; ++i) {
            const float a = alphas[i];
            const float v = a + transLDS[i * LBL + tid];
            if (v > mx) { mx = v; arg = i; }     // strict '>' => first max, as jnp.argmax
            maxA = fmaxf(maxA, a);
        }
        bt[(size_t)t * LBL + tid] = (unsigned char)arg;

        eaLDS[tid] = expf(alphas[tid] - maxA);
        if (t + 4 < TLEN)
            __builtin_prefetch(emit + (size_t)(t + 4) * LBL + tid, 0, 3); // global_prefetch_b8
        __syncthreads();

        // ---- pass 2: s[j] = ea . E[:,j] -----------------------------------
#ifdef HAVE_WMMA_F32X4
        v8f c0 = {};
        v8f c1 = {};
        #pragma unroll
        for (int kt = 0; kt < 32; ++kt) {
            const int kb = kt * 4 + half * 2;
            v2f a;
            a.x = eaLDS[kb + 0];
            a.y = eaLDS[kb + 1];
            if (n != 0) { a.x = 0.0f; a.y = 0.0f; }  // only row M=0 is live
            c0 = __builtin_amdgcn_wmma_f32_16x16x4_f32(
                     false, a, false, Breg[0][kt], (short)0, c0, false, false);
            c1 = __builtin_amdgcn_wmma_f32_16x16x4_f32(
                     false, a, false, Breg[1][kt], (short)0, c1, false, false);
        }
        if (lane < 16) {                    // D row M=0 lives in VGPR0, lanes 0-15
            sBuf[(wave * 2 + 0) * 16 + lane] = c0[0];
            sBuf[(wave * 2 + 1) * 16 + lane] = c1[0];
        }
#else
        {   // VALU fallback (only if the f32x4 WMMA builtin is absent)
            float s = 0.0f;
            for (int i = 0; i < LBL; ++i)
                s += eaLDS[i] * expf(transLDS[i * LBL + tid]);
            sBuf[tid] = s;
        }
#endif
        __syncthreads();

        const float na = maxA + logf(sBuf[tid]) + emit[(size_t)t * LBL + tid];
        alphas[tid] = na;
        __syncthreads();
    }

    // ---- final: score + best last state -----------------------------------
    const float fin = alphas[tid] + transLDS[tid * LBL + STOPT];
    sBuf[tid] = fin;
    __syncthreads();
    if (tid == 0) {
        float m = -__builtin_inff();
        int   bl = 0;
        for (int j = 0; j < LBL; ++j) {
            const float f = sBuf[j];
            if (f > m) { m = f; bl = j; }
        }
        float sum = 0.0f;
        for (int j = 0; j < LBL; ++j) sum += expf(sBuf[j] - m);
        out[0]      = expf(m + logf(sum));   // exp(logsumexp(final)), matches ref
        bestLast[0] = bl;
    }
}

// ---------------------------------------------------------------------------
// Backtrack: inherently sequential pointer chase through L2-resident u8 table.
// bestPath[T-1] = best_last ; bestPath[t-1] = bt[t][bestPath[t]]
// ---------------------------------------------------------------------------
__global__ void crf_backtrack(const unsigned char* __restrict__ bt,
                              const int* __restrict__ bestLast,
                              float* __restrict__ out)
{
    if (threadIdx.x != 0 || blockIdx.x != 0) return;
    int state = bestLast[0];
    out[1 + (TLEN - 1)] = (float)state;
    for (int t = TLEN - 1; t >= 1; --t) {
        state = (int)bt[(size_t)t * LBL + state];
        out[1 + (t - 1)] = (float)state;
    }
}

extern "C" void kernel_launch(void* const* d_in, const int* in_sizes, int n_in,
                              void* d_out, int out_size, void* d_ws, size_t ws_size,
                              hipStream_t stream)
{
    const float* emit  = (const float*)d_in[0];   // (T, L) f32
    const float* trans = (const float*)d_in[1];   // (L, L) f32
    float* out = (float*)d_out;                   // [score, bestPath(T)] as f32

    unsigned char* bt = (unsigned char*)d_ws;                       // T*L bytes
    int* bestLast = (int*)((char*)d_ws + (size_t)TLEN * LBL);       // +4 bytes

    crf_forward<<<1, 128, 0, stream>>>(emit, trans, out, bt, bestLast);
    crf_backtrack<<<1, 32, 0, stream>>>(bt, bestLast, out);
}